// HEncoder_2499670966883
// MI455X (gfx1250) — compile-verified
//
#include <hip/hip_runtime.h>

typedef float v2f __attribute__((ext_vector_type(2)));
typedef float v4f __attribute__((ext_vector_type(4)));
typedef float v8f __attribute__((ext_vector_type(8)));

#define N_ROWS 50000
#define K_NBR  32
#define D_DIM  128
#define XS_STRIDE 130   // 128 + 2 pad floats: rows land on distinct LDS banks, keeps 8B align

// ---------------------------------------------------------------------------
// Kernel 1: h = x @ W^T + b   via V_WMMA_F32_16X16X4_F32
// grid = N/16 blocks, 256 threads (8 waves). Block owns a 16-row M tile,
// wave w owns the 16-column N tile [16w, 16w+16) of D=128.
// ---------------------------------------------------------------------------
__global__ __launch_bounds__(256) void gemm_h_kernel(const float* __restrict__ x,
                                                     const float* __restrict__ W,
                                                     const float* __restrict__ bias,
                                                     float* __restrict__ h) {
    __shared__ float xs[16 * XS_STRIDE];

    const int tid  = threadIdx.x;
    const int wave = tid >> 5;
    const int lane = tid & 31;
    const int m0   = blockIdx.x * 16;

    // Cooperative load of x tile [16 x 128] into LDS (2048 floats / 256 thr = 2 v4f each)
    {
        const v4f* xg = reinterpret_cast<const v4f*>(x + (size_t)m0 * D_DIM);
#pragma unroll
        for (int i = 0; i < 2; ++i) {
            const int f   = tid + i * 256;   // v4f index in [0, 512)
            const int row = f >> 5;          // 32 v4f per row
            const int c4  = f & 31;
            v4f v = xg[row * 32 + c4];
            float* dst = &xs[row * XS_STRIDE + c4 * 4];
            dst[0] = v.x; dst[1] = v.y; dst[2] = v.z; dst[3] = v.w;
        }
    }
    __syncthreads();

    const int n0 = wave * 16;

    // A frag (16x4 f32): lanes 0-15 -> M=lane, K={k0,k0+1}; lanes 16-31 -> M=lane-16, K={k0+2,k0+3}
    const int arow  = (lane < 16) ? lane : (lane - 16);
    const int koffA = (lane < 16) ? 0 : 2;
    // B frag (4x16 f32) = W^T tile: VGPR0 lanes0-15 K=k0, lanes16-31 K=k0+1; VGPR1 K=k0+2 / k0+3
    const int ncolB  = n0 + ((lane < 16) ? lane : (lane - 16));
    const int koffB0 = (lane < 16) ? 0 : 1;
    const float* wrow = W + (size_t)ncolB * D_DIM;   // W is [out,in] row-major; B[k][n] = W[n][k]

    v8f acc = {};

#pragma unroll 8
    for (int k0 = 0; k0 < D_DIM; k0 += 4) {
        v2f a;
        const float* ap = &xs[arow * XS_STRIDE + k0 + koffA];
        a.x = ap[0];
        a.y = ap[1];

        v2f bb;
        bb.x = wrow[k0 + koffB0];
        bb.y = wrow[k0 + koffB0 + 2];

        // 8 args: (neg_a, A, neg_b, B, c_mod, C, reuse_a, reuse_b)
        acc = __builtin_amdgcn_wmma_f32_16x16x4_f32(false, a, false, bb,
                                                    (short)0, acc, false, false);
    }

    // C/D layout: VGPR i -> M = i + (lane>=16 ? 8 : 0), N = lane & 15
    const float bval  = bias[n0 + (lane & 15)];
    const int   mbase = m0 + ((lane < 16) ? 0 : 8);
    const int   ncol  = n0 + (lane & 15);
#pragma unroll
    for (int i = 0; i < 8; ++i) {
        h[(size_t)(mbase + i) * D_DIM + ncol] = acc[i] + bval;
    }
}

// ---------------------------------------------------------------------------
// Kernel 2: compact valid neighbors to the front, gather rows of h.
// One wave per source row n. Lane k holds mask bit k and index k; ballot
// builds the wave-uniform validity mask, deg = popcount. Two uniform loops:
// copies for j<deg (j-th set bit peeled, idx fetched by shuffle), zero fill
// after. 512B rows move as one v4f (b128) per lane, fully coalesced.
// Output stream (845 MB >> 192 MB L2) is stored NON-TEMPORAL so h (25.6 MB)
// stays L2-resident for the gather reads.
// ---------------------------------------------------------------------------
__global__ __launch_bounds__(256) void gather_kernel(const float* __restrict__ h,
                                                     const int* __restrict__ nbr_idx,
                                                     const int* __restrict__ nbr_mask,
                                                     float* __restrict__ out) {
    const int tid  = threadIdx.x;
    const int lane = tid & 31;
    const int wave = tid >> 5;
    const int n    = blockIdx.x * 8 + wave;
    if (n >= N_ROWS) return;

    // lane k holds mask/idx slot k of row n (coalesced 128B reads)
    const int mv   = nbr_mask[(size_t)n * K_NBR + lane];
    const int idxl = nbr_idx [(size_t)n * K_NBR + lane];
    const unsigned m = __builtin_amdgcn_ballot_w32(mv != 0);   // wave32 ballot
    const int deg = __popc(m);                                  // wave-uniform

    v4f* dst = reinterpret_cast<v4f*>(out + (size_t)n * K_NBR * D_DIM);

    unsigned mm = m;   // uniform across the wave
    int j = 0;
    for (; j < deg; ++j) {
        const int p = __ffs(mm) - 1;           // position of j-th valid neighbor
        mm &= mm - 1;
        const int idx = __shfl(idxl, p, 32);   // broadcast idx from lane p
        const v4f v =
            reinterpret_cast<const v4f*>(h + (size_t)idx * D_DIM)[lane];
        __builtin_nontemporal_store(v, &dst[j * 32 + lane]);
    }
    const v4f zero4 = {0.f, 0.f, 0.f, 0.f};
    for (; j < K_NBR; ++j) {
        __builtin_nontemporal_store(zero4, &dst[j * 32 + lane]);
    }
}

// ---------------------------------------------------------------------------
extern "C" void kernel_launch(void* const* d_in, const int* in_sizes, int n_in,
                              void* d_out, int out_size, void* d_ws, size_t ws_size,
                              hipStream_t stream) {
    const float* x        = (const float*)d_in[0];
    const int*   nbr_idx  = (const int*)d_in[1];
    const int*   nbr_mask = (const int*)d_in[2];
    const float* W        = (const float*)d_in[3];
    const float* b        = (const float*)d_in[4];

    float* h     = (float*)d_out;                       // [N, D]
    float* h_nbr = h + (size_t)N_ROWS * D_DIM;          // [N, K, D]

    gemm_h_kernel<<<N_ROWS / 16, 256, 0, stream>>>(x, W, b, h);
    gather_kernel<<<(N_ROWS + 7) / 8, 256, 0, stream>>>(h, nbr_idx, nbr_mask, h_nbr);
}